// TVNet_84035330113895
// MI455X (gfx1250) — compile-verified
//
#include <hip/hip_runtime.h>
#include <hip/hip_bf16.h>

typedef __attribute__((ext_vector_type(2))) float v2f;
typedef __attribute__((ext_vector_type(8))) float v8f;

#define NN   256
#define BATCH 16
#define CH    3
#define CKC   6
#define HW   65536           // 256*256
#define PI_F 3.14159265358979323846f

// ---------------------------------------------------------------- utilities
__global__ void zero_kernel(float* __restrict__ p, long n) {
  long i = (long)blockIdx.x * blockDim.x + threadIdx.x;
  if (i < n) p[i] = 0.0f;
}

// DFT matrix F[j,k] = exp(-2*pi*i*j*k/N): Fr = cos, Fi = -sin (stored as sinf of -ang)
__global__ void dft_kernel(float* __restrict__ Fr, float* __restrict__ Fi) {
  int idx = blockIdx.x * blockDim.x + threadIdx.x;
  if (idx >= HW) return;
  int kq = idx & 255, j = idx >> 8;
  int m = (j * kq) & 255;                      // exact phase reduction mod N
  float ang = -2.0f * PI_F * (float)m * (1.0f / 256.0f);
  float s, c;
  sincosf(ang, &s, &c);
  Fr[idx] = c;
  Fi[idx] = s;
}

// kerH (C,5,5) = sum_o rho[ck] * autocorr(K[ck]) with +1 at center
__global__ void kerh_kernel(const float* __restrict__ K, const float* __restrict__ rho,
                            float* __restrict__ kerH) {
  int t = threadIdx.x;
  if (t >= CH * 25) return;
  int c = t / 25, pos = t % 25, y = pos / 5, xq = pos % 5;
  float acc = 0.0f;
  for (int o = 0; o < 2; ++o) {
    int ck = c * 2 + o;
    const float* Kc = K + ck * 9;
    float s = 0.0f;
    for (int dy = 0; dy < 3; ++dy)
      for (int dx = 0; dx < 3; ++dx) {
        int ay = y + dy - 2, ax = xq + dx - 2;
        if (ay >= 0 && ay < 3 && ax >= 0 && ax < 3)
          s += Kc[dy * 3 + dx] * Kc[ay * 3 + ax];
      }
    acc += rho[ck] * s;
  }
  if (y == 2 && xq == 2) acc += 1.0f;
  kerH[t] = acc;
}

// invOTF[c,u,v] = conj(H)/|H|^2 where H = sum_{i,j} kerH[c,i,j] e^{-2pi i (u(i-2)+v(j-2))/N}
__global__ void otf_kernel(const float* __restrict__ kerH, float* __restrict__ invOTF) {
  int idx = blockIdx.x * blockDim.x + threadIdx.x;
  if (idx >= CH * HW) return;
  int vq = idx & 255, uq = (idx >> 8) & 255, c = idx >> 16;
  float hr = 0.0f, hi = 0.0f;
  for (int i = 0; i < 5; ++i)
    for (int j = 0; j < 5; ++j) {
      float w = kerH[c * 25 + i * 5 + j];
      int ph = ((uq * (i - 2) + vq * (j - 2)) % 256 + 256) & 255;
      float ang = -2.0f * PI_F * (float)ph * (1.0f / 256.0f);
      float s, cc;
      sincosf(ang, &s, &cc);
      hr += w * cc;
      hi += w * s;
    }
  float d = 1.0f / (hr * hr + hi * hi);
  invOTF[2 * idx]     = hr * d;
  invOTF[2 * idx + 1] = -hi * d;
}

// w1 = x + sum_o dwconv(rho*flip(K), p - mu)   (fused adjoint conv, groups summed)
__global__ void w1_kernel(const float* __restrict__ x, const float* __restrict__ p,
                          const float* __restrict__ mu, const float* __restrict__ K,
                          const float* __restrict__ rho, float* __restrict__ w1) {
  int idx = blockIdx.x * blockDim.x + threadIdx.x;
  if (idx >= BATCH * CH * HW) return;
  int xq = idx & 255, yq = (idx >> 8) & 255;
  int c = (idx >> 16) % CH, b = idx / (CH * HW);
  float acc = x[idx];
  for (int o = 0; o < 2; ++o) {
    int ck = c * 2 + o;
    const float* Kc = K + ck * 9;
    float r = rho[ck];
    long base = ((long)(b * CKC + ck)) * HW;
    for (int e = 0; e < 3; ++e) {
      int yy = yq + 1 - e;
      if (yy < 0 || yy > 255) continue;
      for (int f = 0; f < 3; ++f) {
        int xx = xq + 1 - f;
        if (xx < 0 || xx > 255) continue;
        long a = base + yy * 256 + xx;
        acc += r * Kc[e * 3 + f] * (p[a] - mu[a]);
      }
    }
  }
  w1[idx] = acc;
}

// (Zr,Zi) *= invOTF (complex)
__global__ void cmul_kernel(float* __restrict__ Zr, float* __restrict__ Zi,
                            const float* __restrict__ invOTF) {
  int idx = blockIdx.x * blockDim.x + threadIdx.x;
  if (idx >= BATCH * CH * HW) return;
  int pix = idx & (HW - 1);
  int c = (idx >> 16) % CH;
  long o = ((long)c * HW + pix) * 2;
  float ir = invOTF[o], ii = invOTF[o + 1];
  float zr = Zr[idx], zi = Zi[idx];
  Zr[idx] = zr * ir - zi * ii;
  Zi[idx] = zr * ii + zi * ir;
}

// fused: c = dwconv(K, repeat(u)); v = c+mu; p = soft(v, beta/rho); mu += c - p
__global__ void update_kernel(const float* __restrict__ u, const float* __restrict__ K,
                              const float* __restrict__ beta, const float* __restrict__ rho,
                              float* __restrict__ p, float* __restrict__ mu) {
  int idx = blockIdx.x * blockDim.x + threadIdx.x;
  if (idx >= BATCH * CKC * HW) return;
  int xq = idx & 255, yq = (idx >> 8) & 255;
  int ck = (idx >> 16) % CKC, b = idx / (CKC * HW);
  int c = ck >> 1;
  const float* Kc = K + ck * 9;
  long ub = ((long)(b * CH + c)) * HW;
  float cacc = 0.0f;
  for (int dy = 0; dy < 3; ++dy) {
    int yy = yq + dy - 1;
    if (yy < 0 || yy > 255) continue;
    for (int dx = 0; dx < 3; ++dx) {
      int xx = xq + dx - 1;
      if (xx < 0 || xx > 255) continue;
      cacc += Kc[dy * 3 + dx] * u[ub + yy * 256 + xx];
    }
  }
  float gm = beta[ck] / rho[ck];
  float m = mu[idx];
  float v = cacc + m;
  float pv = copysignf(fmaxf(fabsf(v) - gm, 0.0f), v);
  p[idx] = pv;
  mu[idx] = m + cacc - pv;
}

// ------------------------------------------------------ WMMA f32 batched GEMM
// C = scale * (A1*B1 + sign2*A2*B2), all matrices 256x256 row-major.
// Per-batch element strides sA/sB/sC (0 => shared operand).
template <int USE2>
__launch_bounds__(128)
__global__ void gemm_pair_kernel(float* __restrict__ Cm,
                                 const float* __restrict__ A1, const float* __restrict__ B1,
                                 const float* __restrict__ A2, const float* __restrict__ B2,
                                 float sign2, float scale, long sA, long sB, long sC) {
  const int batch = blockIdx.z;
  const float* a1 = A1 + (long)batch * sA;
  const float* b1 = B1 + (long)batch * sB;
  const float* a2 = USE2 ? (A2 + (long)batch * sA) : nullptr;
  const float* b2 = USE2 ? (B2 + (long)batch * sB) : nullptr;
  float* cc = Cm + (long)batch * sC;

  const int wave = threadIdx.x >> 5;
  const int lane = threadIdx.x & 31;
  const int llo = lane & 15;
  const int lhi = lane >> 4;
  const int M0 = blockIdx.y * 64 + (wave >> 1) * 32;
  const int N0 = blockIdx.x * 64 + (wave & 1) * 32;

  v8f acc00 = {}, acc01 = {}, acc10 = {}, acc11 = {};

  for (int k0 = 0; k0 < NN; k0 += 4) {
    const int k = k0 + 2 * lhi;
    // A fragments: 16x4, lanes 0-15 hold K=k0..k0+1, lanes 16-31 hold K=k0+2..k0+3
    v2f a1m0 = *(const v2f*)(a1 + (long)(M0 + llo) * NN + k);
    v2f a1m1 = *(const v2f*)(a1 + (long)(M0 + 16 + llo) * NN + k);
    // B fragments: 4x16, row-striped across lanes
    v2f b1n0, b1n1;
    b1n0.x = b1[(long)k * NN + N0 + llo];
    b1n0.y = b1[(long)(k + 1) * NN + N0 + llo];
    b1n1.x = b1[(long)k * NN + N0 + 16 + llo];
    b1n1.y = b1[(long)(k + 1) * NN + N0 + 16 + llo];

    __builtin_prefetch(b1 + (long)(k + 8) * NN + N0 + llo, 0, 1);

    acc00 = __builtin_amdgcn_wmma_f32_16x16x4_f32(false, a1m0, false, b1n0, (short)0, acc00, false, false);
    acc01 = __builtin_amdgcn_wmma_f32_16x16x4_f32(false, a1m0, false, b1n1, (short)0, acc01, false, false);
    acc10 = __builtin_amdgcn_wmma_f32_16x16x4_f32(false, a1m1, false, b1n0, (short)0, acc10, false, false);
    acc11 = __builtin_amdgcn_wmma_f32_16x16x4_f32(false, a1m1, false, b1n1, (short)0, acc11, false, false);

    if (USE2) {
      v2f a2m0 = *(const v2f*)(a2 + (long)(M0 + llo) * NN + k);
      v2f a2m1 = *(const v2f*)(a2 + (long)(M0 + 16 + llo) * NN + k);
      a2m0 *= sign2;   // f32 WMMA has no A/B negate modifier; fold sign here
      a2m1 *= sign2;
      v2f b2n0, b2n1;
      b2n0.x = b2[(long)k * NN + N0 + llo];
      b2n0.y = b2[(long)(k + 1) * NN + N0 + llo];
      b2n1.x = b2[(long)k * NN + N0 + 16 + llo];
      b2n1.y = b2[(long)(k + 1) * NN + N0 + 16 + llo];
      acc00 = __builtin_amdgcn_wmma_f32_16x16x4_f32(false, a2m0, false, b2n0, (short)0, acc00, false, false);
      acc01 = __builtin_amdgcn_wmma_f32_16x16x4_f32(false, a2m0, false, b2n1, (short)0, acc01, false, false);
      acc10 = __builtin_amdgcn_wmma_f32_16x16x4_f32(false, a2m1, false, b2n0, (short)0, acc10, false, false);
      acc11 = __builtin_amdgcn_wmma_f32_16x16x4_f32(false, a2m1, false, b2n1, (short)0, acc11, false, false);
    }
  }

  // C/D layout: VGPR i -> rows M0+i (lanes 0-15) / M0+8+i (lanes 16-31)
#pragma unroll
  for (int i = 0; i < 8; ++i) {
    int r = M0 + 8 * lhi + i;
    cc[(long)r * NN + N0 + llo]             = acc00[i] * scale;
    cc[(long)r * NN + N0 + 16 + llo]        = acc01[i] * scale;
    cc[(long)(r + 16) * NN + N0 + llo]      = acc10[i] * scale;
    cc[(long)(r + 16) * NN + N0 + 16 + llo] = acc11[i] * scale;
  }
}

// ------------------------------------------------------------------ launcher
extern "C" void kernel_launch(void* const* d_in, const int* in_sizes, int n_in,
                              void* d_out, int out_size, void* d_ws, size_t ws_size,
                              hipStream_t stream) {
  (void)in_sizes; (void)n_in; (void)out_size; (void)ws_size;
  const float* x    = (const float*)d_in[0];
  const float* kerK = (const float*)d_in[1];
  const float* beta = (const float*)d_in[2];
  const float* rho  = (const float*)d_in[3];
  float* out = (float*)d_out;

  const long IMG = (long)BATCH * CH * HW;   // 48 images
  float* ws     = (float*)d_ws;
  float* Fr     = ws;                       // 65536
  float* Fi     = Fr + HW;                  // 65536
  float* kerH   = Fi + HW;                  // 128 (75 used)
  float* invOTF = kerH + 128;               // 2*3*HW
  float* w1     = invOTF + 2L * CH * HW;    // 48*HW
  float* Tr     = w1 + IMG;
  float* Ti     = Tr + IMG;
  float* Zr     = Ti + IMG;
  float* Zi     = Zr + IMG;
  float* p      = Zi + IMG;                 // 96*HW
  float* mu     = p + (long)BATCH * CKC * HW;

  dft_kernel<<<HW / 256, 256, 0, stream>>>(Fr, Fi);
  {
    long n = 2L * BATCH * CKC * HW;         // p and mu are contiguous
    zero_kernel<<<(int)((n + 255) / 256), 256, 0, stream>>>(p, n);
  }

  dim3 gblk(128, 1, 1);
  dim3 ggrid(NN / 64, NN / 64, BATCH * CH);
  const long sI = HW;
  const int ptBlocks = (int)(IMG / 256);
  const int upBlocks = (int)(((long)BATCH * CKC * HW) / 256);

  for (int i = 0; i < 10; ++i) {
    const float* Ki = kerK + (long)i * CKC * 9;
    const float* bi = beta + (long)i * CKC;
    const float* ri = rho  + (long)i * CKC;

    kerh_kernel<<<1, 128, 0, stream>>>(Ki, ri, kerH);
    otf_kernel<<<(CH * HW) / 256, 256, 0, stream>>>(kerH, invOTF);
    w1_kernel<<<ptBlocks, 256, 0, stream>>>(x, p, mu, Ki, ri, w1);

    // forward 2-D FFT:  T = F * w1 ;  Z = T * F
    gemm_pair_kernel<0><<<ggrid, gblk, 0, stream>>>(Tr, Fr, w1, nullptr, nullptr, 1.f, 1.f, 0, sI, sI);
    gemm_pair_kernel<0><<<ggrid, gblk, 0, stream>>>(Ti, Fi, w1, nullptr, nullptr, 1.f, 1.f, 0, sI, sI);
    gemm_pair_kernel<1><<<ggrid, gblk, 0, stream>>>(Zr, Tr, Fr, Ti, Fi, -1.f, 1.f, sI, 0, sI);
    gemm_pair_kernel<1><<<ggrid, gblk, 0, stream>>>(Zi, Tr, Fi, Ti, Fr, +1.f, 1.f, sI, 0, sI);

    cmul_kernel<<<ptBlocks, 256, 0, stream>>>(Zr, Zi, invOTF);

    // inverse 2-D FFT:  S = conj(F) * Z ;  u = Re(S * conj(F)) / N^2  -> d_out
    gemm_pair_kernel<1><<<ggrid, gblk, 0, stream>>>(Tr, Fr, Zr, Fi, Zi, +1.f, 1.f, 0, sI, sI);
    gemm_pair_kernel<1><<<ggrid, gblk, 0, stream>>>(Ti, Fr, Zi, Fi, Zr, -1.f, 1.f, 0, sI, sI);
    gemm_pair_kernel<1><<<ggrid, gblk, 0, stream>>>(out, Tr, Fr, Ti, Fi, +1.f, 1.0f / (float)HW, sI, 0, sI);

    update_kernel<<<upBlocks, 256, 0, stream>>>(out, Ki, bi, ri, p, mu);
  }
}